// LearnedWLSEmbeddings_2671469658617
// MI455X (gfx1250) — compile-verified
//
#include <hip/hip_runtime.h>

typedef float v2f __attribute__((ext_vector_type(2)));
typedef float v8f __attribute__((ext_vector_type(8)));

#define ANS 65   // stride for An (32x64), padded -> conflict-free column gathers
#define MST 66   // stride for M (64x64): bank = (2r+c)%64, distinct per row-lane
#define YST 34   // stride for Y (64x32): bank = (34r+c)%64, distinct per row-lane

__device__ __forceinline__ void lin4_relu(const float* __restrict__ W,
                                          const float* __restrict__ bb,
                                          const float* in, float* out) {
#pragma unroll
  for (int o = 0; o < 4; ++o) {
    float a = bb[o];
#pragma unroll
    for (int i = 0; i < 4; ++i) a += W[o * 4 + i] * in[i];
    out[o] = a > 0.f ? a : 0.f;
  }
}

__global__ __launch_bounds__(32)
void wls_kernel(const float* __restrict__ x, const float* __restrict__ Z,
                const float* __restrict__ AW1, const float* __restrict__ Ab1,
                const float* __restrict__ AW2, const float* __restrict__ Ab2,
                const float* __restrict__ AW3, const float* __restrict__ Ab3,
                const float* __restrict__ AW4, const float* __restrict__ Ab4,
                const float* __restrict__ bW1, const float* __restrict__ bb1,
                const float* __restrict__ bW2, const float* __restrict__ bb2,
                const float* __restrict__ bW3, const float* __restrict__ bb3,
                const float* __restrict__ bW4, const float* __restrict__ bb4,
                const float* __restrict__ dW1, const float* __restrict__ db1,
                const float* __restrict__ dW2, const float* __restrict__ db2,
                const float* __restrict__ dW3, const float* __restrict__ db3,
                const float* __restrict__ dW4, const float* __restrict__ db4,
                float* __restrict__ out) {
  __shared__ float sAn[32 * ANS];   // An: [row t][feat]
  __shared__ float sM[64 * MST];    // diag(Z)+S, destroyed by Gauss-Jordan
  __shared__ float sY[64 * YST];    // Y1 -> X1 = term2
  __shared__ float sA12[4 * 64];
  __shared__ float sG[4 * 64];      // A11inv @ A12
  __shared__ float sH[4 * 32];      // A11inv @ AbT
  __shared__ float sAb[32 * 4];
  __shared__ float sR[32];
  __shared__ float sA11[16];
  __shared__ float sA11i[16];
  __shared__ float sT1[4 * 32];     // term1
  __shared__ float sE[68];
  __shared__ float sZ[64];
  __shared__ float sD1[16];
  __shared__ float sD2[16];
  __shared__ float sD3[4];

  const int lane = threadIdx.x;
  const int b = blockIdx.x;

  sZ[lane] = Z[lane];
  sZ[lane + 32] = Z[lane + 32];

  // ---- Stage 1: per-row tiny MLPs (lane t = row t) ----
  const float* xr = x + ((size_t)b * 32 + lane) * 4;
  float xin[4] = {xr[0], xr[1], xr[2], xr[3]};
  {
    float h1[4], h2[4], h3[4];
    lin4_relu(AW1, Ab1, xin, h1);
    lin4_relu(AW2, Ab2, h1, h2);
    lin4_relu(AW3, Ab3, h2, h3);
#pragma unroll 8
    for (int o = 0; o < 64; ++o) {
      float a = Ab4[o];
#pragma unroll
      for (int i = 0; i < 4; ++i) a += AW4[o * 4 + i] * h3[i];
      sAn[lane * ANS + o] = a;
    }
    float g1[4], g2[4], g3[4];
    lin4_relu(bW1, bb1, xin, g1);
    lin4_relu(bW2, bb2, g1, g2);
    lin4_relu(bW3, bb3, g2, g3);
    float rn = bb4[0];
#pragma unroll
    for (int i = 0; i < 4; ++i) rn += bW4[i] * g3[i];
    sR[lane] = xin[0] + rn;                 // r_base + r_net
    sAb[lane * 4 + 0] = -xin[1];
    sAb[lane * 4 + 1] = -xin[2];
    sAb[lane * 4 + 2] = -xin[3];
    sAb[lane * 4 + 3] = 1.0f;
  }
  __syncthreads();

  // ---- Stage 2: A11 = AbT@Ab and its 4x4 inverse ----
  if (lane < 16) {
    int i = lane >> 2, j = lane & 3;
    float s = 0.f;
    for (int t = 0; t < 32; ++t) s += sAb[t * 4 + i] * sAb[t * 4 + j];
    sA11[lane] = s;
  }
  __syncthreads();
  if (lane == 0) {
    const float* m = sA11;
    float inv[16];
    inv[0]  =  m[5]*m[10]*m[15] - m[5]*m[11]*m[14] - m[9]*m[6]*m[15] + m[9]*m[7]*m[14] + m[13]*m[6]*m[11] - m[13]*m[7]*m[10];
    inv[4]  = -m[4]*m[10]*m[15] + m[4]*m[11]*m[14] + m[8]*m[6]*m[15] - m[8]*m[7]*m[14] - m[12]*m[6]*m[11] + m[12]*m[7]*m[10];
    inv[8]  =  m[4]*m[9]*m[15]  - m[4]*m[11]*m[13] - m[8]*m[5]*m[15] + m[8]*m[7]*m[13] + m[12]*m[5]*m[11] - m[12]*m[7]*m[9];
    inv[12] = -m[4]*m[9]*m[14]  + m[4]*m[10]*m[13] + m[8]*m[5]*m[14] - m[8]*m[6]*m[13] - m[12]*m[5]*m[10] + m[12]*m[6]*m[9];
    inv[1]  = -m[1]*m[10]*m[15] + m[1]*m[11]*m[14] + m[9]*m[2]*m[15] - m[9]*m[3]*m[14] - m[13]*m[2]*m[11] + m[13]*m[3]*m[10];
    inv[5]  =  m[0]*m[10]*m[15] - m[0]*m[11]*m[14] - m[8]*m[2]*m[15] + m[8]*m[3]*m[14] + m[12]*m[2]*m[11] - m[12]*m[3]*m[10];
    inv[9]  = -m[0]*m[9]*m[15]  + m[0]*m[11]*m[13] + m[8]*m[1]*m[15] - m[8]*m[3]*m[13] - m[12]*m[1]*m[11] + m[12]*m[3]*m[9];
    inv[13] =  m[0]*m[9]*m[14]  - m[0]*m[10]*m[13] - m[8]*m[1]*m[14] + m[8]*m[2]*m[13] + m[12]*m[1]*m[10] - m[12]*m[2]*m[9];
    inv[2]  =  m[1]*m[6]*m[15]  - m[1]*m[7]*m[14]  - m[5]*m[2]*m[15] + m[5]*m[3]*m[14] + m[13]*m[2]*m[7]  - m[13]*m[3]*m[6];
    inv[6]  = -m[0]*m[6]*m[15]  + m[0]*m[7]*m[14]  + m[4]*m[2]*m[15] - m[4]*m[3]*m[14] - m[12]*m[2]*m[7]  + m[12]*m[3]*m[6];
    inv[10] =  m[0]*m[5]*m[15]  - m[0]*m[7]*m[13]  - m[4]*m[1]*m[15] + m[4]*m[3]*m[13] + m[12]*m[1]*m[7]  - m[12]*m[3]*m[5];
    inv[14] = -m[0]*m[5]*m[14]  + m[0]*m[6]*m[13]  + m[4]*m[1]*m[14] - m[4]*m[2]*m[13] - m[12]*m[1]*m[6]  + m[12]*m[2]*m[5];
    inv[3]  = -m[1]*m[6]*m[11]  + m[1]*m[7]*m[10]  + m[5]*m[2]*m[11] - m[5]*m[3]*m[10] - m[9]*m[2]*m[7]   + m[9]*m[3]*m[6];
    inv[7]  =  m[0]*m[6]*m[11]  - m[0]*m[7]*m[10]  - m[4]*m[2]*m[11] + m[4]*m[3]*m[10] + m[8]*m[2]*m[7]   - m[8]*m[3]*m[6];
    inv[11] = -m[0]*m[5]*m[11]  + m[0]*m[7]*m[9]   + m[4]*m[1]*m[11] - m[4]*m[3]*m[9]  - m[8]*m[1]*m[7]   + m[8]*m[3]*m[5];
    inv[15] =  m[0]*m[5]*m[10]  - m[0]*m[6]*m[9]   - m[4]*m[1]*m[10] + m[4]*m[2]*m[9]  + m[8]*m[1]*m[6]   - m[8]*m[2]*m[5];
    float det = m[0]*inv[0] + m[1]*inv[4] + m[2]*inv[8] + m[3]*inv[12];
    float rdet = 1.0f / det;
#pragma unroll
    for (int i = 0; i < 16; ++i) sA11i[i] = inv[i] * rdet;
  }
  __syncthreads();

  // ---- Stage 3: A12 = AbT @ An (4x64); A21 == A12^T ----
#pragma unroll
  for (int e = 0; e < 8; ++e) {
    int idx = lane * 8 + e;
    int k = idx >> 6, col = idx & 63;
    float s = 0.f;
    for (int t = 0; t < 32; ++t) s += sAb[t * 4 + k] * sAn[t * ANS + col];
    sA12[k * 64 + col] = s;
  }
  __syncthreads();

  // ---- Stage 4: G = A11inv@A12 (4x64), H = A11inv@AbT (4x32) ----
#pragma unroll
  for (int cc = 0; cc < 2; ++cc) {
    int col = lane + 32 * cc;
#pragma unroll
    for (int i = 0; i < 4; ++i) {
      float s = 0.f;
#pragma unroll
      for (int k = 0; k < 4; ++k) s += sA11i[i * 4 + k] * sA12[k * 64 + col];
      sG[i * 64 + col] = s;
    }
  }
#pragma unroll
  for (int k = 0; k < 4; ++k) {
    float s = 0.f;
#pragma unroll
    for (int j = 0; j < 4; ++j) s += sA11i[k * 4 + j] * sAb[lane * 4 + j];
    sH[k * 32 + lane] = s;
  }
  __syncthreads();

  // ---- Stage 5: Y1 = AnT - A12^T @ H (64x32), column-per-lane ----
  for (int i = 0; i < 64; ++i) {
    float s = sAn[lane * ANS + i];
#pragma unroll
    for (int k = 0; k < 4; ++k) s -= sA12[k * 64 + i] * sH[k * 32 + lane];
    sY[i * YST + lane] = s;
  }

  // ---- Stage 6: S = AnT@An - A12^T@G via V_WMMA_F32_16X16X4_F32 ----
  {
    const int half = lane >> 4;     // K half: lanes 0-15 -> K{0,1}, 16-31 -> K{2,3}
    const int mrow = lane & 15;     // M (for A) / N (for B) index
#pragma unroll
    for (int I = 0; I < 4; ++I) {
#pragma unroll
      for (int J = 0; J < 4; ++J) {
        v8f acc = {};
#pragma unroll
        for (int kb = 0; kb < 8; ++kb) {          // contraction over 32 rows
          int r0 = 4 * kb + 2 * half;
          v2f a, bv;
          a.x  = sAn[(r0 + 0) * ANS + 16 * I + mrow];
          a.y  = sAn[(r0 + 1) * ANS + 16 * I + mrow];
          bv.x = sAn[(r0 + 0) * ANS + 16 * J + mrow];
          bv.y = sAn[(r0 + 1) * ANS + 16 * J + mrow];
          acc = __builtin_amdgcn_wmma_f32_16x16x4_f32(false, a, false, bv,
                                                      (short)0, acc, false, false);
        }
        {                                          // -A12^T @ G correction (K=4)
          int k0 = 2 * half;
          v2f a, bv;
          a.x  = -sA12[(k0 + 0) * 64 + 16 * I + mrow];
          a.y  = -sA12[(k0 + 1) * 64 + 16 * I + mrow];
          bv.x =  sG[(k0 + 0) * 64 + 16 * J + mrow];
          bv.y =  sG[(k0 + 1) * 64 + 16 * J + mrow];
          acc = __builtin_amdgcn_wmma_f32_16x16x4_f32(false, a, false, bv,
                                                      (short)0, acc, false, false);
        }
        // branch-free tile store; diag(Z) added in a post-pass
        const int rbase = (16 * I + 8 * half) * MST + 16 * J + mrow;
#pragma unroll
        for (int v = 0; v < 8; ++v) sM[rbase + v * MST] = acc[v];
      }
    }
  }
  __syncthreads();
  // ---- diag(Z) add: M = diag(Z) + S, one lane per row, no divergence ----
  sM[lane * MST + lane] += sZ[lane];
  sM[(lane + 32) * MST + (lane + 32)] += sZ[lane + 32];
  __syncthreads();

  // ---- Stage 7: Gauss-Jordan solve M X = Y1 (SPD, no pivoting). Y -> term2 ----
  for (int p = 0; p < 64; ++p) {
    float invp = 1.0f / sM[p * MST + p];
    sM[p * MST + lane]      *= invp;
    sM[p * MST + lane + 32] *= invp;
    sY[p * YST + lane]      *= invp;
    __syncthreads();
#pragma unroll
    for (int rr = 0; rr < 2; ++rr) {
      int r = lane + 32 * rr;
      if (r != p) {
        float f = sM[r * MST + p];
#pragma unroll 4
        for (int c = p + 1; c < 64; ++c) sM[r * MST + c] -= f * sM[p * MST + c];
#pragma unroll 8
        for (int c = 0; c < 32; ++c)     sY[r * YST + c] -= f * sY[p * YST + c];
      }
    }
    __syncthreads();
  }

  // ---- Stage 8: term1 = H - G @ X1 (4x32), column-per-lane ----
#pragma unroll
  for (int k = 0; k < 4; ++k) {
    float s = sH[k * 32 + lane];
    for (int i = 0; i < 64; ++i) s -= sG[k * 64 + i] * sY[i * YST + lane];
    sT1[k * 32 + lane] = s;
  }
  __syncthreads();

  // ---- Stage 9: embedding e = A_inv @ (r_base + r_net), 68 entries ----
  {
    int idx = lane;
    float s = 0.f;
    if (idx < 4) { for (int t = 0; t < 32; ++t) s += sT1[idx * 32 + t] * sR[t]; }
    else { int i = idx - 4; for (int t = 0; t < 32; ++t) s += sY[i * YST + t] * sR[t]; }
    sE[idx] = s;
  }
  {
    int i = lane + 28;                 // idx = lane+32 -> i = idx-4
    float s = 0.f;
    for (int t = 0; t < 32; ++t) s += sY[i * YST + t] * sR[t];
    sE[lane + 32] = s;
  }
  if (lane < 4) {
    int i = 60 + lane;                 // idx = 64+lane
    float s = 0.f;
    for (int t = 0; t < 32; ++t) s += sY[i * YST + t] * sR[t];
    sE[64 + lane] = s;
  }
  __syncthreads();

  // ---- Stage 10: decoder MLP + skip ----
  if (lane < 16) {
    float a = db1[lane];
    for (int i = 0; i < 68; ++i) a += dW1[lane * 68 + i] * sE[i];
    sD1[lane] = a > 0.f ? a : 0.f;
  }
  __syncthreads();
  if (lane < 16) {
    float a = db2[lane];
#pragma unroll
    for (int i = 0; i < 16; ++i) a += dW2[lane * 16 + i] * sD1[i];
    sD2[lane] = a > 0.f ? a : 0.f;
  }
  __syncthreads();
  if (lane < 4) {
    float a = db3[lane];
#pragma unroll
    for (int i = 0; i < 16; ++i) a += dW3[lane * 16 + i] * sD2[i];
    sD3[lane] = a > 0.f ? a : 0.f;
  }
  __syncthreads();
  if (lane < 4) {
    float a = db4[lane];
#pragma unroll
    for (int i = 0; i < 4; ++i) a += dW4[lane * 4 + i] * sD3[i];
    out[(size_t)b * 4 + lane] = a + sE[lane];
  }
}

extern "C" void kernel_launch(void* const* d_in, const int* in_sizes, int n_in,
                              void* d_out, int out_size, void* d_ws, size_t ws_size,
                              hipStream_t stream) {
  (void)d_ws; (void)ws_size; (void)out_size;
  const float* x = (const float*)d_in[0];
  const float* Z = (const float*)d_in[2];
  const int B = in_sizes[0] / (32 * 4);

  const float *AW1, *Ab1, *AW2, *Ab2, *AW3, *Ab3, *AW4, *Ab4;
  const float *bW1, *bb1, *bW2, *bb2, *bW3, *bb3, *bW4, *bb4;
  const float *dW1, *db1, *dW2, *db2, *dW3, *db3, *dW4, *db4;

  if (n_in >= 27) {  // pytree leaves: each W/b its own input
    AW1 = (const float*)d_in[3];  Ab1 = (const float*)d_in[4];
    AW2 = (const float*)d_in[5];  Ab2 = (const float*)d_in[6];
    AW3 = (const float*)d_in[7];  Ab3 = (const float*)d_in[8];
    AW4 = (const float*)d_in[9];  Ab4 = (const float*)d_in[10];
    bW1 = (const float*)d_in[11]; bb1 = (const float*)d_in[12];
    bW2 = (const float*)d_in[13]; bb2 = (const float*)d_in[14];
    bW3 = (const float*)d_in[15]; bb3 = (const float*)d_in[16];
    bW4 = (const float*)d_in[17]; bb4 = (const float*)d_in[18];
    dW1 = (const float*)d_in[19]; db1 = (const float*)d_in[20];
    dW2 = (const float*)d_in[21]; db2 = (const float*)d_in[22];
    dW3 = (const float*)d_in[23]; db3 = (const float*)d_in[24];
    dW4 = (const float*)d_in[25]; db4 = (const float*)d_in[26];
  } else {           // each param tuple flattened into one blob
    const float* A = (const float*)d_in[3];
    AW1 = A + 0;   Ab1 = A + 16;  AW2 = A + 20;  Ab2 = A + 36;
    AW3 = A + 40;  Ab3 = A + 56;  AW4 = A + 60;  Ab4 = A + 316;   // 64x4 W, 64 b
    const float* Bp = (const float*)d_in[4];
    bW1 = Bp + 0;  bb1 = Bp + 16; bW2 = Bp + 20; bb2 = Bp + 36;
    bW3 = Bp + 40; bb3 = Bp + 56; bW4 = Bp + 60; bb4 = Bp + 64;   // 1x4 W, 1 b
    const float* D = (const float*)d_in[5];
    dW1 = D + 0;    db1 = D + 1088; dW2 = D + 1104; db2 = D + 1360;
    dW3 = D + 1376; db3 = D + 1440; dW4 = D + 1444; db4 = D + 1460;
  }

  wls_kernel<<<B, 32, 0, stream>>>(x, Z,
      AW1, Ab1, AW2, Ab2, AW3, Ab3, AW4, Ab4,
      bW1, bb1, bW2, bb2, bW3, bb3, bW4, bb4,
      dW1, db1, dW2, db2, dW3, db3, dW4, db4,
      (float*)d_out);
}